// MattingSolver_22462678958638
// MI455X (gfx1250) — compile-verified
//
#include <hip/hip_runtime.h>

// MI455X / gfx1250 CG solver for the 5-point grid Laplacian (matting solver).
// Strategy: COO -> 5-band ELL via closed-form row pointers (no scatter/atomics),
// whole working set (~32MB) resident in the 192MB L2, float4 (b128) vector
// kernels to minimize VMEM instructions per byte, f32 WMMA for dot products.

#define HGRID 1024
#define WGRID 1024
#define NELEM (HGRID * WGRID)
#define BLOCKT 256
#define CG_STEPS 50
#define NBLK_VEC (NELEM / (BLOCKT * 4)) // 1024 blocks, one float4 per thread
#define NBLK_DOT (NELEM / 4096)         // 256 blocks; 8 waves * 8 tiles * 64

typedef float v2f __attribute__((ext_vector_type(2)));
typedef float v8f __attribute__((ext_vector_type(8)));

// ---------------------------------------------------------------------------
// Build 5-band ELL from row-sorted COO using closed-form row offsets, and
// initialize x=0, r=b, p=b.
// Band order: 0 -> col=i-W, 1 -> i-1, 2 -> i (diag), 3 -> i+1, 4 -> i+W.
// ---------------------------------------------------------------------------
__global__ void build_bands_init(const int* __restrict__ cols,
                                 const float* __restrict__ vals,
                                 const float* __restrict__ bvec,
                                 float* __restrict__ bands,
                                 float* __restrict__ rv,
                                 float* __restrict__ pv,
                                 float* __restrict__ x) {
  int i = blockIdx.x * BLOCKT + threadIdx.x;
  int gr = i >> 10;            // grid row
  int gc = i & (WGRID - 1);    // grid col
  int cnt = 1 + (gc > 0) + (gc < WGRID - 1) + (gr > 0) + (gr < HGRID - 1);
  // entries in all grid-rows before gr: row0 = 4W-2, middle rows = 5W-2
  int pref = (gr > 0) ? (4 * WGRID - 2) + (gr - 1) * (5 * WGRID - 2) : 0;
  // entries in this grid-row for columns < gc
  int part = gc                              // diagonals
           + ((gc > 0) ? (gc - 1) : 0)       // left-neighbor entries (c'>0)
           + gc                              // right-neighbor entries (c'<W-1)
           + ((gr > 0) ? gc : 0)             // up-neighbor entries
           + ((gr < HGRID - 1) ? gc : 0);    // down-neighbor entries
  int s = pref + part;

  float b0 = 0.f, b1 = 0.f, b2 = 0.f, b3 = 0.f, b4 = 0.f;
  for (int e = 0; e < cnt; ++e) {
    int col = cols[s + e];
    float v = vals[s + e];
    int d = col - i;
    if (d == -WGRID)      b0 = v;
    else if (d == -1)     b1 = v;
    else if (d == 0)      b2 = v;
    else if (d == 1)      b3 = v;
    else                  b4 = v;   // d == +WGRID
  }
  bands[0ull * NELEM + i] = b0;
  bands[1ull * NELEM + i] = b1;
  bands[2ull * NELEM + i] = b2;
  bands[3ull * NELEM + i] = b3;
  bands[4ull * NELEM + i] = b4;

  float bi = bvec[i];
  rv[i] = bi;
  pv[i] = bi;
  x[i] = 0.f;
}

// ---------------------------------------------------------------------------
// Banded SpMV, float4 per thread. Row stride (1024) is a multiple of 4, so
// up/down neighbor vectors stay 16B-aligned; clamped addresses only occur
// where the corresponding band coefficient is zero. Left/right neighbors of
// interior vector lanes come from the center float4 in registers; only the
// two edge scalars (p[i-1], p[i+4]) need extra loads.
// ---------------------------------------------------------------------------
__global__ void spmv_banded(const float* __restrict__ bands,
                            const float* __restrict__ p,
                            float* __restrict__ Ap) {
  int i = (blockIdx.x * BLOCKT + threadIdx.x) * 4;

  float4 c  = *(const float4*)(p + i);
  int iU = (i >= WGRID) ? i - WGRID : i;           // coeff b0 == 0 when clamped
  int iD = (i + WGRID < NELEM) ? i + WGRID : i;    // coeff b4 == 0 when clamped
  float4 up = *(const float4*)(p + iU);
  float4 dn = *(const float4*)(p + iD);
  float pl = (i > 0) ? p[i - 1] : 0.f;             // coeff b1.x == 0 at i==0
  float pr = (i + 4 < NELEM) ? p[i + 4] : 0.f;     // coeff b3.w == 0 at end

  float4 b0 = *(const float4*)(bands + 0ull * NELEM + i);
  float4 b1 = *(const float4*)(bands + 1ull * NELEM + i);
  float4 b2 = *(const float4*)(bands + 2ull * NELEM + i);
  float4 b3 = *(const float4*)(bands + 3ull * NELEM + i);
  float4 b4 = *(const float4*)(bands + 4ull * NELEM + i);

  float4 y;
  y.x = fmaf(b0.x, up.x, fmaf(b1.x, pl,  fmaf(b3.x, c.y, fmaf(b4.x, dn.x, b2.x * c.x))));
  y.y = fmaf(b0.y, up.y, fmaf(b1.y, c.x, fmaf(b3.y, c.z, fmaf(b4.y, dn.y, b2.y * c.y))));
  y.z = fmaf(b0.z, up.z, fmaf(b1.z, c.y, fmaf(b3.z, c.w, fmaf(b4.z, dn.z, b2.z * c.z))));
  y.w = fmaf(b0.w, up.w, fmaf(b1.w, c.z, fmaf(b3.w, pr,  fmaf(b4.w, dn.w, b2.w * c.w))));
  *(float4*)(Ap + i) = y;
}

// ---------------------------------------------------------------------------
// dot(u,v) partials via V_WMMA_F32_16X16X4_F32.
// Per 64-element tile: A[m][k] = u[base+4m+k] (16x4), B[k][n] = v[base+4n+k]
// (4x16)  =>  C[m][n] += dot(u-seg m, v-seg n); diagonal C[m][m] accumulates
// the wanted products. Off-diagonal lanes are discarded at extraction.
// Diagonal of C (16x16 f32): lanes 0-7 at VGPR=lane, lanes 24-31 at VGPR=lane-24.
// ---------------------------------------------------------------------------
__global__ void dot_wmma(const float* __restrict__ u,
                         const float* __restrict__ v,
                         float* __restrict__ partials) {
  int tid = threadIdx.x;
  int wave = tid >> 5;
  int lane = tid & 31;
  int m = lane & 15;
  int g = lane >> 4;
  int waveBase = blockIdx.x * 4096 + wave * 512;

  v8f c = {};
#pragma unroll
  for (int t = 0; t < 8; ++t) {
    int base = waveBase + t * 64;
    int ia = base + m * 4 + 2 * g;
    v2f a;  a.x = u[ia];     a.y = u[ia + 1];
    int ib = base + m * 4 + g;
    v2f bb; bb.x = v[ib];    bb.y = v[ib + 2];
    c = __builtin_amdgcn_wmma_f32_16x16x4_f32(false, a, false, bb,
                                              (short)0, c, false, false);
  }

  // extract this lane's diagonal element (if any)
  int sel = (lane < 8) ? lane : ((lane >= 24) ? lane - 24 : -1);
  float d = 0.f;
  d = (sel == 0) ? c[0] : d;
  d = (sel == 1) ? c[1] : d;
  d = (sel == 2) ? c[2] : d;
  d = (sel == 3) ? c[3] : d;
  d = (sel == 4) ? c[4] : d;
  d = (sel == 5) ? c[5] : d;
  d = (sel == 6) ? c[6] : d;
  d = (sel == 7) ? c[7] : d;

  // wave32 butterfly reduce
  for (int off = 16; off >= 1; off >>= 1) d += __shfl_xor(d, off, 32);

  __shared__ float wsum[BLOCKT / 32];
  if (lane == 0) wsum[wave] = d;
  __syncthreads();
  if (tid == 0) {
    float s = 0.f;
#pragma unroll
    for (int w = 0; w < BLOCKT / 32; ++w) s += wsum[w];
    partials[blockIdx.x] = s;
  }
}

// deterministic in-block reduction of NBLK_DOT (=256) partials
__device__ __forceinline__ float reduce_partials(const float* __restrict__ partials,
                                                 float* red, int tid) {
  red[tid] = partials[tid];
  __syncthreads();
  for (int s = BLOCKT / 2; s > 0; s >>= 1) {
    if (tid < s) red[tid] += red[tid + s];
    __syncthreads();
  }
  return red[0];
}

// alpha = rs/pAp ; x += alpha*p ; r -= alpha*Ap   (b128 streams)
__global__ void xr_update(const float* __restrict__ partials,
                          const float* __restrict__ rs_old,
                          float* __restrict__ x,
                          float* __restrict__ r,
                          const float* __restrict__ p,
                          const float* __restrict__ Ap) {
  __shared__ float red[BLOCKT];
  int tid = threadIdx.x;
  float pAp = reduce_partials(partials, red, tid);
  float alpha = rs_old[0] / pAp;
  int i = (blockIdx.x * BLOCKT + tid) * 4;

  float4 pv  = *(const float4*)(p + i);
  float4 apv = *(const float4*)(Ap + i);
  float4 xv  = *(const float4*)(x + i);
  float4 rv  = *(const float4*)(r + i);
  xv.x = fmaf(alpha, pv.x, xv.x);  rv.x = fmaf(-alpha, apv.x, rv.x);
  xv.y = fmaf(alpha, pv.y, xv.y);  rv.y = fmaf(-alpha, apv.y, rv.y);
  xv.z = fmaf(alpha, pv.z, xv.z);  rv.z = fmaf(-alpha, apv.z, rv.z);
  xv.w = fmaf(alpha, pv.w, xv.w);  rv.w = fmaf(-alpha, apv.w, rv.w);
  *(float4*)(x + i) = xv;
  *(float4*)(r + i) = rv;
}

// beta = rs_new/rs_old ; p = r + beta*p ; rs_next = rs_new (single writer)
__global__ void p_update(const float* __restrict__ partials,
                         const float* __restrict__ rs_old,
                         float* __restrict__ rs_next,
                         const float* __restrict__ r,
                         float* __restrict__ p) {
  __shared__ float red[BLOCKT];
  int tid = threadIdx.x;
  float rs_new = reduce_partials(partials, red, tid);
  float beta = rs_new / rs_old[0];
  int i = (blockIdx.x * BLOCKT + tid) * 4;

  float4 pv = *(const float4*)(p + i);
  float4 rv = *(const float4*)(r + i);
  pv.x = fmaf(beta, pv.x, rv.x);
  pv.y = fmaf(beta, pv.y, rv.y);
  pv.z = fmaf(beta, pv.z, rv.z);
  pv.w = fmaf(beta, pv.w, rv.w);
  *(float4*)(p + i) = pv;

  if (blockIdx.x == 0 && tid == 0) rs_next[0] = rs_new;
}

// single-block finalize of the initial rs0 = dot(b,b)
__global__ void store_rs(const float* __restrict__ partials,
                         float* __restrict__ rs0) {
  __shared__ float red[BLOCKT];
  int tid = threadIdx.x;
  float s = reduce_partials(partials, red, tid);
  if (tid == 0) rs0[0] = s;
}

// ---------------------------------------------------------------------------
extern "C" void kernel_launch(void* const* d_in, const int* in_sizes, int n_in,
                              void* d_out, int out_size, void* d_ws, size_t ws_size,
                              hipStream_t stream) {
  (void)in_sizes; (void)n_in; (void)out_size; (void)ws_size;
  // inputs: rows (unused; row offsets are closed-form), cols, vals, b
  const int*   cols = (const int*)d_in[1];
  const float* vals = (const float*)d_in[2];
  const float* b    = (const float*)d_in[3];
  float* x = (float*)d_out;

  float* ws      = (float*)d_ws;
  float* bands   = ws;                      // 5n floats (20 MB)
  float* r       = ws + 5ull * NELEM;       // n
  float* p       = ws + 6ull * NELEM;       // n
  float* Ap      = ws + 7ull * NELEM;       // n
  float* partial = ws + 8ull * NELEM;       // NBLK_DOT (256)
  float* rs_hist = partial + BLOCKT;        // CG_STEPS+1 scalars

  build_bands_init<<<NELEM / BLOCKT, BLOCKT, 0, stream>>>(cols, vals, b,
                                                          bands, r, p, x);
  // rs0 = dot(r,r)
  dot_wmma<<<NBLK_DOT, BLOCKT, 0, stream>>>(r, r, partial);
  store_rs<<<1, BLOCKT, 0, stream>>>(partial, rs_hist);

  for (int t = 0; t < CG_STEPS; ++t) {
    spmv_banded<<<NBLK_VEC, BLOCKT, 0, stream>>>(bands, p, Ap);
    dot_wmma<<<NBLK_DOT, BLOCKT, 0, stream>>>(p, Ap, partial);
    xr_update<<<NBLK_VEC, BLOCKT, 0, stream>>>(partial, rs_hist + t,
                                               x, r, p, Ap);
    dot_wmma<<<NBLK_DOT, BLOCKT, 0, stream>>>(r, r, partial);
    p_update<<<NBLK_VEC, BLOCKT, 0, stream>>>(partial, rs_hist + t,
                                              rs_hist + t + 1, r, p);
  }
}